// LSTM_CRF_70720931496031
// MI455X (gfx1250) — compile-verified
//
#include <hip/hip_runtime.h>
#include <hip/hip_bf16.h>
#include <math.h>

// Problem constants (match reference)
#define BB 32
#define TT 512
#define DD 768
#define HH 256
#define KK 32
#define GG (4 * HH)   // 1024 gates per direction
#define MM (BB * TT)  // 16384 GEMM rows

typedef float v2f __attribute__((ext_vector_type(2)));
typedef float v8f __attribute__((ext_vector_type(8)));

// ---------------------------------------------------------------------------
// WMMA f32 GEMM:  Out[r][n] = sum_k A[r][k] * Bm[n][k] + bias[n]
// A: M x K row-major (lda), Bm: N x K row-major (ldb), Out: M x N (ldo).
// Tiles: BM=64 x BN (template), KC=32. 256 threads = 8 waves.
// Requires M%64==0, N%BN==0, K%32==0 (true for all uses here).
// ---------------------------------------------------------------------------
template <int BN>
__global__ __launch_bounds__(256) void wmma_gemm_nt(
    const float* __restrict__ A, const float* __restrict__ Bm,
    const float* __restrict__ bias, float* __restrict__ Out,
    int M, int N, int K, int lda, int ldb, int ldo)
{
    constexpr int BM = 64, KC = 32, LSTR = KC + 2;   // LDS row stride 34 -> bank-conflict free
    __shared__ float As[BM * LSTR];
    __shared__ float Bs[BN * LSTR];

    const int tid  = threadIdx.x;
    const int wave = tid >> 5;
    const int lane = tid & 31;
    const int lr   = lane & 15;          // row index inside a 16-wide fragment
    const int kh   = (lane >> 4) << 1;   // K sub-offset: 0 (lanes 0-15) or 2 (lanes 16-31)

    const int mTileBase = blockIdx.x * BM;
    const int nTileBase = blockIdx.y * BN;

    const int mSub = (wave & 3) * 16;              // 4 M-tiles across waves
    constexpr int NTW = BN / 32;                   // 16-wide N-tiles per wave
    const int nSubBase = (wave >> 2) * NTW * 16;   // 2 wave groups in N

    v8f acc[NTW] = {};

    for (int k0 = 0; k0 < K; k0 += KC) {
        __syncthreads();
        // Stage A tile (BM x KC) into LDS, float4 global loads
        for (int i = tid; i < BM * (KC / 4); i += 256) {
            int r  = i >> 3;
            int c4 = (i & 7) << 2;
            float4 v = *(const float4*)(A + (size_t)(mTileBase + r) * lda + k0 + c4);
            float* dst = &As[r * LSTR + c4];
            dst[0] = v.x; dst[1] = v.y; dst[2] = v.z; dst[3] = v.w;
        }
        // Stage B tile (BN x KC) into LDS
        for (int i = tid; i < BN * (KC / 4); i += 256) {
            int r  = i >> 3;
            int c4 = (i & 7) << 2;
            float4 v = *(const float4*)(Bm + (size_t)(nTileBase + r) * ldb + k0 + c4);
            float* dst = &Bs[r * LSTR + c4];
            dst[0] = v.x; dst[1] = v.y; dst[2] = v.z; dst[3] = v.w;
        }
        __syncthreads();

        #pragma unroll
        for (int kk = 0; kk < KC; kk += 4) {
            // A fragment: 16x4 f32: lane holds (M=lr, K = kk+kh .. kk+kh+1)
            v2f a;
            a.x = As[(mSub + lr) * LSTR + kk + kh];
            a.y = As[(mSub + lr) * LSTR + kk + kh + 1];
            #pragma unroll
            for (int nt = 0; nt < NTW; ++nt) {
                // B fragment: 4x16 f32: lane holds (N=lr, K = kk+kh .. kk+kh+1)
                const int nr = nSubBase + nt * 16 + lr;
                v2f b;
                b.x = Bs[nr * LSTR + kk + kh];
                b.y = Bs[nr * LSTR + kk + kh + 1];
                acc[nt] = __builtin_amdgcn_wmma_f32_16x16x4_f32(
                    false, a, false, b, (short)0, acc[nt], false, false);
            }
        }
    }

    // C/D layout: element r of acc <-> (M = mSub + r + 8*(lane>=16), N = lr)
    const int rowOff = (lane >> 4) << 3;  // 0 or 8
    for (int nt = 0; nt < NTW; ++nt) {
        const int col = nTileBase + nSubBase + nt * 16 + lr;
        const float bv = bias[col];
        #pragma unroll
        for (int r = 0; r < 8; ++r) {
            const int row = mTileBase + mSub + rowOff + r;
            Out[(size_t)row * ldo + col] = acc[nt][r] + bv;
        }
    }
}

// ---------------------------------------------------------------------------
// LSTM recurrence. One workgroup per (direction, batch): 64 blocks x 256 thr.
// Thread tid owns hidden unit tid: gate rows tid, tid+H, tid+2H, tid+3H of
// W_hh (4H x H row-major). h lives in LDS; c in a register. pre layout:
// [dir][b][t][4H]. hcat layout: [b][t][2H] (fwd in [0,H), bwd in [H,2H)).
// ---------------------------------------------------------------------------
__global__ __launch_bounds__(256) void lstm_recurrent(
    const float* __restrict__ pre,
    const float* __restrict__ Whh_f, const float* __restrict__ Whh_b,
    const int* __restrict__ lengths,
    float* __restrict__ hcat)
{
    const int tid = threadIdx.x;
    const int dir = blockIdx.x >> 5;
    const int b   = blockIdx.x & 31;
    const float* W = dir ? Whh_b : Whh_f;
    const int len = lengths[b];

    __shared__ float hs[HH];
    hs[tid] = 0.0f;
    float c = 0.0f;

    const float4* Wi = (const float4*)(W + (size_t)(tid         ) * HH);
    const float4* Wf = (const float4*)(W + (size_t)(tid +     HH) * HH);
    const float4* Wg = (const float4*)(W + (size_t)(tid + 2 * HH) * HH);
    const float4* Wo = (const float4*)(W + (size_t)(tid + 3 * HH) * HH);
    const float4* h4 = (const float4*)hs;

    const float* preD = pre + ((size_t)dir * BB + b) * TT * GG;

    __syncthreads();
    for (int t = 0; t < TT; ++t) {
        // forward: process time t; backward: process x_rev[t] = x[rev_idx(t)]
        const int tsrc = (dir == 0) ? t : ((t < len) ? (len - 1 - t) : t);
        const float* prow = preD + (size_t)tsrc * GG;
        float ai = prow[tid];
        float af = prow[tid + HH];
        float ag = prow[tid + 2 * HH];
        float ao = prow[tid + 3 * HH];
        #pragma unroll 4
        for (int k = 0; k < HH / 4; ++k) {
            float4 hv = h4[k];
            float4 wi = Wi[k], wf = Wf[k], wg = Wg[k], wo = Wo[k];
            ai += hv.x * wi.x + hv.y * wi.y + hv.z * wi.z + hv.w * wi.w;
            af += hv.x * wf.x + hv.y * wf.y + hv.z * wf.z + hv.w * wf.w;
            ag += hv.x * wg.x + hv.y * wg.y + hv.z * wg.z + hv.w * wg.w;
            ao += hv.x * wo.x + hv.y * wo.y + hv.z * wo.z + hv.w * wo.w;
        }
        const float ig = 1.0f / (1.0f + expf(-ai));
        const float fg = 1.0f / (1.0f + expf(-af));
        const float og = 1.0f / (1.0f + expf(-ao));
        c = fg * c + ig * tanhf(ag);
        const float h = og * tanhf(c);

        __syncthreads();              // everyone done reading old h
        hs[tid] = h;
        // backward result is un-reversed: store at tsrc (== t for forward)
        hcat[((size_t)b * TT + tsrc) * (2 * HH) + dir * HH + tid] = h;
        __syncthreads();
    }
}

// ---------------------------------------------------------------------------
// CRF negative log-likelihood. One block, 32 waves: wave = batch, lane = tag.
// alpha held one element per lane; LSE over source tags via __shfl broadcast.
// ---------------------------------------------------------------------------
__global__ __launch_bounds__(1024) void crf_nll_kernel(
    const float* __restrict__ em, const int* __restrict__ targets,
    const int* __restrict__ lengths,
    const float* __restrict__ start_t, const float* __restrict__ end_t,
    const float* __restrict__ trans, float* __restrict__ loss_out)
{
    const int tid = threadIdx.x;
    const int b = tid >> 5;
    const int k = tid & 31;
    const int len = lengths[b];

    __shared__ float tr_s[KK * KK];
    __shared__ float red[BB];
    for (int i = tid; i < KK * KK; i += 1024) tr_s[i] = trans[i];
    __syncthreads();

    const float* emb = em + (size_t)b * TT * KK;
    float alpha = start_t[k] + emb[k];

    for (int t = 1; t < TT; ++t) {
        const float e = emb[t * KK + k];
        float mx = -INFINITY;
        #pragma unroll
        for (int j = 0; j < KK; ++j) {
            const float aj = __shfl(alpha, j, 32);
            mx = fmaxf(mx, aj + tr_s[j * KK + k]);
        }
        float s = 0.0f;
        #pragma unroll
        for (int j = 0; j < KK; ++j) {
            const float aj = __shfl(alpha, j, 32);
            s += expf(aj + tr_s[j * KK + k] - mx);
        }
        const float nxt = e + mx + logf(s);
        alpha = (t < len) ? nxt : alpha;
    }

    // logZ = LSE_k(alpha + end)
    float v = alpha + end_t[k];
    float mx = v;
    for (int off = 16; off; off >>= 1) mx = fmaxf(mx, __shfl_xor(mx, off, 32));
    float s = expf(v - mx);
    for (int off = 16; off; off >>= 1) s += __shfl_xor(s, off, 32);
    const float logZ = mx + logf(s);

    // Numerator: lanes split the time axis (512/32 = 16 iters each)
    const int* tgt = targets + (size_t)b * TT;
    float part = 0.0f;
    for (int t = k; t < TT; t += 32) {
        if (t == 0) {
            part += start_t[tgt[0]] + emb[tgt[0]];
        } else if (t < len) {
            part += tr_s[tgt[t - 1] * KK + tgt[t]] + emb[(size_t)t * KK + tgt[t]];
        }
    }
    for (int off = 16; off; off >>= 1) part += __shfl_xor(part, off, 32);
    const float num = part + end_t[tgt[len - 1]];

    if (k == 0) red[b] = num - logZ;
    __syncthreads();
    if (tid == 0) {
        float acc = 0.0f;
        for (int i = 0; i < BB; ++i) acc += red[i];
        loss_out[0] = -acc / (float)BB;
    }
}

// ---------------------------------------------------------------------------
// Viterbi. One block, wave = batch, lane = tag. History as u8 in workspace.
// ---------------------------------------------------------------------------
__global__ __launch_bounds__(1024) void viterbi_kernel(
    const float* __restrict__ em, const int* __restrict__ lengths,
    const float* __restrict__ start_t, const float* __restrict__ end_t,
    const float* __restrict__ trans,
    unsigned char* __restrict__ hist,   // [T-1][B][K]
    float* __restrict__ preds_out)      // [B][T] as float
{
    const int tid = threadIdx.x;
    const int b = tid >> 5;
    const int k = tid & 31;
    const int len = lengths[b];

    __shared__ float tr_s[KK * KK];
    for (int i = tid; i < KK * KK; i += 1024) tr_s[i] = trans[i];
    __syncthreads();

    const float* emb = em + (size_t)b * TT * KK;
    float score = start_t[k] + emb[k];

    for (int t = 1; t < TT; ++t) {
        const float e = emb[t * KK + k];
        float best = -INFINITY;
        int bj = 0;
        #pragma unroll
        for (int j = 0; j < KK; ++j) {
            const float sj = __shfl(score, j, 32);
            const float vj = sj + tr_s[j * KK + k];
            if (vj > best) { best = vj; bj = j; }   // first max wins (argmax)
        }
        const bool m = (t < len);
        hist[((size_t)(t - 1) * BB + b) * KK + k] = (unsigned char)(m ? bj : k);
        score = m ? (best + e) : score;
    }

    // last = argmax_k(score + end), lowest index on ties
    float bm = score + end_t[k];
    int bi = k;
    for (int off = 16; off; off >>= 1) {
        const float ov = __shfl_xor(bm, off, 32);
        const int   oi = __shfl_xor(bi, off, 32);
        if (ov > bm || (ov == bm && oi < bi)) { bm = ov; bi = oi; }
    }

    __threadfence();  // make this wave's hist stores visible before backtrack
    if (k == 0) {
        int tag = bi;
        preds_out[(size_t)b * TT + (TT - 1)] = ((TT - 1) < len) ? (float)tag : 0.0f;
        for (int t = TT - 2; t >= 0; --t) {
            tag = hist[((size_t)t * BB + b) * KK + tag];
            preds_out[(size_t)b * TT + t] = (t < len) ? (float)tag : 0.0f;
        }
    }
}

// ---------------------------------------------------------------------------
extern "C" void kernel_launch(void* const* d_in, const int* in_sizes, int n_in,
                              void* d_out, int out_size, void* d_ws, size_t ws_size,
                              hipStream_t stream)
{
    const float* x       = (const float*)d_in[0];
    const int*   lengths = (const int*)  d_in[1];
    // d_in[2] = mask (recomputed from lengths; unused)
    const int*   targets = (const int*)  d_in[3];
    const float* W_ih_f  = (const float*)d_in[4];
    const float* W_hh_f  = (const float*)d_in[5];
    const float* b_f     = (const float*)d_in[6];
    const float* W_ih_b  = (const float*)d_in[7];
    const float* W_hh_b  = (const float*)d_in[8];
    const float* b_b     = (const float*)d_in[9];
    const float* W_clf   = (const float*)d_in[10];
    const float* b_clf   = (const float*)d_in[11];
    const float* start_t = (const float*)d_in[12];
    const float* end_t   = (const float*)d_in[13];
    const float* trans   = (const float*)d_in[14];

    // Workspace carve-up
    char* ws = (char*)d_ws;
    const size_t preBytes  = (size_t)2 * BB * TT * GG * sizeof(float);   // 128 MB
    const size_t hcatBytes = (size_t)BB * TT * 2 * HH * sizeof(float);   //  32 MB
    const size_t emBytes   = (size_t)BB * TT * KK * sizeof(float);       //   2 MB
    float* pre  = (float*)ws;
    float* hcat = (float*)(ws + preBytes);
    float* em   = (float*)(ws + preBytes + hcatBytes);
    unsigned char* hist = (unsigned char*)(ws + preBytes + hcatBytes + emBytes);

    dim3 blk(256);

    // Input projections (WMMA f32): pre[dir] = x @ W_ih^T + b
    wmma_gemm_nt<128><<<dim3(MM / 64, GG / 128), blk, 0, stream>>>(
        x, W_ih_f, b_f, pre, MM, GG, DD, DD, DD, GG);
    wmma_gemm_nt<128><<<dim3(MM / 64, GG / 128), blk, 0, stream>>>(
        x, W_ih_b, b_b, pre + (size_t)BB * TT * GG, MM, GG, DD, DD, DD, GG);

    // Bidirectional LSTM recurrence: 64 independent (dir, batch) chains
    lstm_recurrent<<<64, blk, 0, stream>>>(pre, W_hh_f, W_hh_b, lengths, hcat);

    // Classifier (WMMA f32): em = hcat @ W_clf^T + b_clf
    wmma_gemm_nt<32><<<dim3(MM / 64, 1), blk, 0, stream>>>(
        hcat, W_clf, b_clf, em, MM, KK, 2 * HH, 2 * HH, 2 * HH, KK);

    // CRF loss -> d_out[0]
    crf_nll_kernel<<<1, 1024, 0, stream>>>(
        em, targets, lengths, start_t, end_t, trans, (float*)d_out);

    // Viterbi decode -> d_out[1 .. 1+B*T)
    viterbi_kernel<<<1, 1024, 0, stream>>>(
        em, lengths, start_t, end_t, trans, hist, (float*)d_out + 1);
}